// GraphConvBlock_45200235823724
// MI455X (gfx1250) — compile-verified
//
#include <hip/hip_runtime.h>

typedef __attribute__((ext_vector_type(2))) float v2f;
typedef __attribute__((ext_vector_type(8))) float v8f;

#define N_NODES 50000
#define N_EDGES 800000
#define FEAT    128

// ---------------------------------------------------------------------------
// Kernel 1: zero the aggregation workspace (d_ws is poisoned by the harness).
// ---------------------------------------------------------------------------
__global__ void __launch_bounds__(256) zero_f4_kernel(float4* __restrict__ p, int n4) {
    int i = blockIdx.x * blockDim.x + threadIdx.x;
    if (i < n4) p[i] = make_float4(0.f, 0.f, 0.f, 0.f);
}

// ---------------------------------------------------------------------------
// Kernel 2: edge scatter-add:  agg[dst[e]][:] += x[src[e]][:]
// One wave32 per edge; lane l handles floats [4l, 4l+4).  The 32-lane float4
// load covers the full 512B source row (coalesced, L2-resident since x is
// 25.6 MB << 192 MB L2).  4 global_atomic_add_f32 per lane.
// ---------------------------------------------------------------------------
__global__ void __launch_bounds__(256) scatter_kernel(
    const float*      __restrict__ x,
    const long long*  __restrict__ src,
    const long long*  __restrict__ dst,
    float*            __restrict__ agg) {
    int gid  = blockIdx.x * blockDim.x + threadIdx.x;
    int edge = gid >> 5;
    int lane = gid & 31;
    if (edge >= N_EDGES) return;

    long long s = src[edge];
    long long d = dst[edge];

    float4 v = *((const float4*)(x + (size_t)s * FEAT) + lane);
    float* ap = agg + (size_t)d * FEAT + lane * 4;
    atomicAdd(ap + 0, v.x);
    atomicAdd(ap + 1, v.y);
    atomicAdd(ap + 2, v.z);
    atomicAdd(ap + 3, v.w);
}

// ---------------------------------------------------------------------------
// Kernel 3: fused dual-GEMM epilogue using V_WMMA_F32_16X16X4_F32 (exact fp32):
//   out = relu( agg @ W_rel^T + x @ W_root^T + b_rel )
// One wave -> one 16x16 output tile.  blockIdx.x = M-tile (16 rows),
// wave id (0..7) = N-tile (16 cols); 8 waves/block share weight-row reuse.
//
// Fragment layouts (ISA 7.12.2, 16x16x4 f32):
//   A (16x4):  lane<16 holds A[m=lane][k0+0..1],  lane>=16 holds A[m][k0+2..3]
//   B (4x16):  lane<16 holds B[k0+0..1][n=lane],  lane>=16 holds B[k0+2..3][n]
//              out = agg @ W^T  =>  B[k][n] = W[n*128 + k]  (contiguous v2f)
//   C/D (16x16): vgpr r, lane l: element (m = r + 8*(l>=16), n = l%16)
// ---------------------------------------------------------------------------
__global__ void __launch_bounds__(256) gemm_wmma_kernel(
    const float* __restrict__ agg,
    const float* __restrict__ x,
    const float* __restrict__ Wrel,
    const float* __restrict__ brel,
    const float* __restrict__ Wroot,
    float*       __restrict__ out) {
    const int lane = threadIdx.x & 31;
    const int wave = threadIdx.x >> 5;      // 0..7 -> N tile
    const int m0   = blockIdx.x * 16;       // M-tile base row (50000 % 16 == 0)
    const int n0   = wave * 16;             // N-tile base col

    const int half = lane >> 4;             // 0: lanes 0-15, 1: lanes 16-31
    const int l16  = lane & 15;

    const float* aggRow   = agg   + (size_t)(m0 + l16) * FEAT + half * 2;
    const float* xRow     = x     + (size_t)(m0 + l16) * FEAT + half * 2;
    const float* wrelRow  = Wrel  + (size_t)(n0 + l16) * FEAT + half * 2;
    const float* wrootRow = Wroot + (size_t)(n0 + l16) * FEAT + half * 2;

    v8f c = {};
    // agg @ W_rel^T
    #pragma unroll
    for (int k = 0; k < FEAT; k += 4) {
        v2f a = *(const v2f*)(aggRow  + k);
        v2f b = *(const v2f*)(wrelRow + k);
        c = __builtin_amdgcn_wmma_f32_16x16x4_f32(false, a, false, b,
                                                  (short)0, c, false, false);
    }
    // + x @ W_root^T (same accumulator)
    #pragma unroll
    for (int k = 0; k < FEAT; k += 4) {
        v2f a = *(const v2f*)(xRow     + k);
        v2f b = *(const v2f*)(wrootRow + k);
        c = __builtin_amdgcn_wmma_f32_16x16x4_f32(false, a, false, b,
                                                  (short)0, c, false, false);
    }

    const float bias = brel[n0 + l16];      // bias indexed by output feature n
    #pragma unroll
    for (int r = 0; r < 8; ++r) {
        float v = c[r] + bias;
        v = v > 0.f ? v : 0.f;              // ReLU
        out[(size_t)(m0 + r + half * 8) * FEAT + n0 + l16] = v;
    }
}

// ---------------------------------------------------------------------------
extern "C" void kernel_launch(void* const* d_in, const int* in_sizes, int n_in,
                              void* d_out, int out_size, void* d_ws, size_t ws_size,
                              hipStream_t stream) {
    const float*     x     = (const float*)d_in[0];
    const long long* eidx  = (const long long*)d_in[1];   // int64 [2, E]
    const float*     Wrel  = (const float*)d_in[2];
    const float*     brel  = (const float*)d_in[3];
    const float*     Wroot = (const float*)d_in[4];
    float*           out   = (float*)d_out;

    const long long* src = eidx;            // row 0
    const long long* dst = eidx + N_EDGES;  // row 1

    float* agg = (float*)d_ws;              // 50000*128*4 = 25.6 MB scratch

    // 1) zero agg
    {
        int n4 = (N_NODES * FEAT) / 4;      // 1.6M float4
        int blocks = (n4 + 255) / 256;
        zero_f4_kernel<<<blocks, 256, 0, stream>>>((float4*)agg, n4);
    }
    // 2) scatter-add over edges (one wave per edge)
    {
        long long threads = (long long)N_EDGES * 32;
        int blocks = (int)((threads + 255) / 256);
        scatter_kernel<<<blocks, 256, 0, stream>>>(x, src, dst, agg);
    }
    // 3) fused dual-GEMM + bias + ReLU via f32 WMMA
    {
        int mtiles = N_NODES / 16;          // 3125
        gemm_wmma_kernel<<<mtiles, 256, 0, stream>>>(agg, x, Wrel, brel, Wroot, out);
    }
}